// Filip_loss_91182155694750
// MI455X (gfx1250) — compile-verified
//
#include <hip/hip_runtime.h>
#include <cstdint>

// FILIP loss for MI455X (gfx1250), wave32 + V_WMMA_F32_16X16X4_F32.
// B=64 batches, T=64 text tokens, I=196 image patches, D=512.
// Stage 1: one block per (x,y) pair -> t2i[x,y], i2t[x,y] (fp32 WMMA GEMM + masked max/mean).
// Stage 2: contrastive exp/log reduction -> (filip_loss, t2i_loss, i2t_loss).

typedef __attribute__((ext_vector_type(2))) float v2f;
typedef __attribute__((ext_vector_type(8))) float v8f;

#define B_  64
#define T_  64
#define I_  196
#define D_  512
#define NT  13                 // ceil(196/16) N-tiles of 16
#define NEGF (-3.402823466e38f)

__global__ __launch_bounds__(128) void filip_sim_kernel(
    const float* __restrict__ vision,        // [B, I, D]
    const float* __restrict__ text,          // [B, T, D]
    const uint8_t* __restrict__ vmask,       // [B, I] (bool)
    const uint8_t* __restrict__ tmask,       // [B, T] (bool)
    float* __restrict__ t2i,                 // [B, B]
    float* __restrict__ i2t)                 // [B, B]
{
    const int x    = blockIdx.x;             // text batch index
    const int y    = blockIdx.y;             // vision batch index
    const int tid  = threadIdx.x;
    const int wid  = tid >> 5;               // wave id 0..3 -> M tile (t rows wid*16..wid*16+15)
    const int lane = tid & 31;
    const int half = lane >> 4;              // 0: lanes 0-15, 1: lanes 16-31
    const int l16  = lane & 15;

    __shared__ float rowmax_s[T_];
    __shared__ float colmax_s[4][NT * 16];

    const float temp = 1.0f / 0.07f;         // exp(log(1/0.07))

    // A row for this lane: text[x, wid*16 + l16, :]
    const float* trow = text + ((size_t)x * T_ + (size_t)(wid * 16 + l16)) * D_;

    // B rows per N tile: vision[y, nt*16 + l16, :]  (clamped so padded cols stay in-bounds)
    const float* brow[NT];
#pragma unroll
    for (int nt = 0; nt < NT; ++nt) {
        int i = nt * 16 + l16;
        if (i > I_ - 1) i = I_ - 1;
        brow[nt] = vision + ((size_t)y * I_ + (size_t)i) * D_;
    }

    v8f acc[NT];
#pragma unroll
    for (int nt = 0; nt < NT; ++nt) {
        v8f z = {0.f, 0.f, 0.f, 0.f, 0.f, 0.f, 0.f, 0.f};
        acc[nt] = z;
    }

    // K loop: chunk of 8. Lanes 0-15 carry K = {kc..kc+3}, lanes 16-31 carry K = {kc+4..kc+7}
    // (dot product is K-permutation invariant), so one b128 load feeds two WMMA steps.
    for (int kc = 0; kc < D_; kc += 8) {
        const int ka = kc + half * 4;
        float4 a4 = *(const float4*)(trow + ka);
        v2f a0 = {a4.x, a4.y};
        v2f a1 = {a4.z, a4.w};
#pragma unroll
        for (int nt = 0; nt < NT; ++nt) {
            float4 b4 = *(const float4*)(brow[nt] + ka);
            v2f b0 = {b4.x, b4.y};
            v2f b1 = {b4.z, b4.w};
            acc[nt] = __builtin_amdgcn_wmma_f32_16x16x4_f32(
                false, a0, false, b0, (short)0, acc[nt], false, false);
            acc[nt] = __builtin_amdgcn_wmma_f32_16x16x4_f32(
                false, a1, false, b1, (short)0, acc[nt], false, false);
        }
    }

    // C/D layout: VGPR r, lanes 0-15 -> row M=r, lanes 16-31 -> row M=r+8; column N = lane%16.
    // Per-lane masks.
    bool vm[NT];
#pragma unroll
    for (int nt = 0; nt < NT; ++nt) {
        int i = nt * 16 + l16;
        vm[nt] = (i < I_) && (vmask[(size_t)y * I_ + (i < I_ ? i : 0)] != 0);
    }
    bool tm[8];
#pragma unroll
    for (int r = 0; r < 8; ++r)
        tm[r] = tmask[(size_t)x * T_ + (size_t)(wid * 16 + half * 8 + r)] != 0;

    // Row max over image patches i (t2i path), reduced across the 16 lanes of each half.
#pragma unroll
    for (int r = 0; r < 8; ++r) {
        float m = NEGF;
#pragma unroll
        for (int nt = 0; nt < NT; ++nt)
            m = fmaxf(m, vm[nt] ? acc[nt][r] * temp : NEGF);
        m = fmaxf(m, __shfl_xor(m, 1, 32));
        m = fmaxf(m, __shfl_xor(m, 2, 32));
        m = fmaxf(m, __shfl_xor(m, 4, 32));
        m = fmaxf(m, __shfl_xor(m, 8, 32));
        if (l16 == 0) rowmax_s[wid * 16 + half * 8 + r] = m;
    }

    // Column max over this wave's 16 text rows (i2t path), combine lane halves.
#pragma unroll
    for (int nt = 0; nt < NT; ++nt) {
        float m = NEGF;
#pragma unroll
        for (int r = 0; r < 8; ++r)
            m = fmaxf(m, tm[r] ? acc[nt][r] * temp : NEGF);
        m = fmaxf(m, __shfl_xor(m, 16, 32));
        if (half == 0) colmax_s[wid][nt * 16 + l16] = m;
    }
    __syncthreads();

    if (tid == 0) {
        // masked mean over text tokens -> t2i[x,y]
        float s = 0.f, c = 0.f;
        for (int t = 0; t < T_; ++t) {
            if (tmask[(size_t)x * T_ + t]) { s += rowmax_s[t]; c += 1.f; }
        }
        t2i[x * B_ + y] = s / fmaxf(c, 1e-6f);
        // cross-wave col max + masked mean over image patches -> i2t[x,y]
        float si = 0.f, ci = 0.f;
        for (int i = 0; i < I_; ++i) {
            if (vmask[(size_t)y * I_ + i]) {
                float m = colmax_s[0][i];
                m = fmaxf(m, colmax_s[1][i]);
                m = fmaxf(m, colmax_s[2][i]);
                m = fmaxf(m, colmax_s[3][i]);
                si += m; ci += 1.f;
            }
        }
        i2t[x * B_ + y] = si / fmaxf(ci, 1e-6f);
    }
}

__global__ __launch_bounds__(64) void filip_loss_kernel(
    const float* __restrict__ t2i, const float* __restrict__ i2t,
    float* __restrict__ out)
{
    __shared__ float lt_s[B_];
    __shared__ float li_s[B_];
    const int tid = threadIdx.x;              // 0..63

    // t2i: thread = row x; denom = sum over y
    float s = 0.f;
    for (int yy = 0; yy < B_; ++yy) s += expf(t2i[tid * B_ + yy]);
    float lt = -logf(expf(t2i[tid * B_ + tid]) + 1e-20f) + logf(s + 1e-20f);

    // i2t: thread = col y; denom = sum over x
    float s2 = 0.f;
    for (int xx = 0; xx < B_; ++xx) s2 += expf(i2t[xx * B_ + tid]);
    float li = -logf(expf(i2t[tid * B_ + tid]) + 1e-20f) + logf(s2 + 1e-20f);

    lt_s[tid] = lt;
    li_s[tid] = li;
    __syncthreads();
    if (tid == 0) {
        float a = 0.f, b = 0.f;
        for (int k = 0; k < B_; ++k) { a += lt_s[k]; b += li_s[k]; }
        a *= (1.0f / B_);
        b *= (1.0f / B_);
        out[0] = 0.5f * (a + b);   // filip_loss
        out[1] = a;                // t2i_loss
        out[2] = b;                // i2t_loss
    }
}

extern "C" void kernel_launch(void* const* d_in, const int* in_sizes, int n_in,
                              void* d_out, int out_size, void* d_ws, size_t ws_size,
                              hipStream_t stream) {
    (void)in_sizes; (void)n_in; (void)out_size; (void)ws_size;
    const float*   vision = (const float*)d_in[0];     // [64,196,512] f32
    const float*   text   = (const float*)d_in[1];     // [64,64,512]  f32
    const uint8_t* vmask  = (const uint8_t*)d_in[2];   // [64,196] bool
    const uint8_t* tmask  = (const uint8_t*)d_in[3];   // [64,64]  bool

    float* t2i = (float*)d_ws;                         // [64,64]
    float* i2t = t2i + B_ * B_;                        // [64,64]

    dim3 grid(B_, B_);
    filip_sim_kernel<<<grid, 128, 0, stream>>>(vision, text, vmask, tmask, t2i, i2t);
    filip_loss_kernel<<<1, 64, 0, stream>>>(t2i, i2t, (float*)d_out);
}